// GCN3_80977313399734
// MI455X (gfx1250) — compile-verified
//
#include <hip/hip_runtime.h>
#include <hip/hip_bf16.h>
#include <cstddef>

typedef __attribute__((ext_vector_type(2))) float v2f;
typedef __attribute__((ext_vector_type(8))) float v8f;

#define D_IN 256
#define D_H  64
#define D_O  47
#define D_OP 48   // padded

// ---------------------------------------------------------------------------
// Degree / normalization
// ---------------------------------------------------------------------------
__global__ void k_deg_init(float* __restrict__ deg, int N) {
    int i = blockIdx.x * blockDim.x + threadIdx.x;
    if (i < N) deg[i] = 1.0f;                       // self-loop weight
}

__global__ void k_deg_acc(const int* __restrict__ dst, float* __restrict__ deg, int E) {
    int e = blockIdx.x * blockDim.x + threadIdx.x;
    if (e < E) atomicAdd(&deg[dst[e]], 1.0f);
}

__global__ void k_deg_rsqrt(float* __restrict__ deg, int N) {
    int i = blockIdx.x * blockDim.x + threadIdx.x;
    if (i < N) deg[i] = rsqrtf(deg[i]);             // deg >= 1 always
}

// ---------------------------------------------------------------------------
// Pack B[K, bcols] (row-major) into WMMA lane order, zero-padding cols>=bcols.
// Bp[(k4*NT + t)*32 + lane] = v2f{ B[kk][col], B[kk+1][col] }
//   kk = 4*k4 + 2*(lane>>4), col = t*16 + (lane&15)
// ---------------------------------------------------------------------------
template <int NT>
__global__ void k_packB(const float* __restrict__ B, v2f* __restrict__ Bp,
                        int K, int bcols) {
    int idx = blockIdx.x * blockDim.x + threadIdx.x;
    int total = (K >> 2) * NT * 32;
    if (idx >= total) return;
    int lane = idx & 31;
    int rest = idx >> 5;
    int t = rest % NT;
    int k4 = rest / NT;
    int half = lane >> 4;
    int l16 = lane & 15;
    int kk = 4 * k4 + 2 * half;
    int col = t * 16 + l16;
    v2f b;
    b.x = (col < bcols) ? B[(size_t)kk * bcols + col] : 0.0f;
    b.y = (col < bcols) ? B[(size_t)(kk + 1) * bcols + col] : 0.0f;
    Bp[idx] = b;
}

// ---------------------------------------------------------------------------
// WMMA fp32 GEMM: C[M, ldc] = A[M, K] @ Bp (pre-packed, pre-padded).
// One wave computes a 16 x (16*NT) output stripe. M must be multiple of 16.
// Inner loop is branch-free: 1 A load + NT B loads + NT v_wmma per K-step.
// ---------------------------------------------------------------------------
template <int NT>
__global__ void k_gemm_wmma(const float* __restrict__ A, const v2f* __restrict__ Bp,
                            float* __restrict__ C, int K, int ldc, int numMtiles) {
    const int wave = blockIdx.x * (blockDim.x >> 5) + (threadIdx.x >> 5);
    if (wave >= numMtiles) return;                  // wave-uniform: EXEC stays all-1s
    const int lane = threadIdx.x & 31;
    const int half = lane >> 4;                     // 0: K=k,k+1   1: K=k+2,k+3
    const int l16  = lane & 15;

    v8f acc[NT];
#pragma unroll
    for (int t = 0; t < NT; ++t) acc[t] = {};

    const size_t row = (size_t)wave * 16 + l16;     // A-matrix row for this lane
    const float* __restrict__ Arow = A + row * (size_t)K + 2 * half;
    const v2f* __restrict__ Bl = Bp + lane;

    const int K4 = K >> 2;
    for (int k4 = 0; k4 < K4; ++k4) {
        v2f a = *(const v2f*)(Arow + 4 * k4);
#pragma unroll
        for (int t = 0; t < NT; ++t) {
            v2f b = Bl[(size_t)(k4 * NT + t) * 32];
            acc[t] = __builtin_amdgcn_wmma_f32_16x16x4_f32(
                false, a, false, b, (short)0, acc[t], false, false);
        }
    }

    // C/D layout: VGPR r -> lanes 0-15: (M=r, N=lane), lanes 16-31: (M=r+8, N=lane-16)
#pragma unroll
    for (int t = 0; t < NT; ++t) {
        const int n = t * 16 + l16;
#pragma unroll
        for (int r = 0; r < 8; ++r) {
            const size_t m = (size_t)wave * 16 + r + half * 8;
            C[m * (size_t)ldc + n] = acc[t][r];
        }
    }
}

// ---------------------------------------------------------------------------
// Self-loop initialization + edge scatter-add (layer 1, width 64)
// ---------------------------------------------------------------------------
__global__ void k_self1(const float* __restrict__ dinv, const float* __restrict__ h,
                        float* __restrict__ acc, int N) {
    int idx = blockIdx.x * blockDim.x + threadIdx.x;
    if (idx < N * D_H) {
        int i = idx >> 6;
        float w = dinv[i] * dinv[i];
        acc[idx] = w * h[idx];
    }
}

__global__ void k_scatter1(const int* __restrict__ src, const int* __restrict__ dst,
                           const float* __restrict__ dinv, const float* __restrict__ h,
                           float* __restrict__ acc, int E) {
    int idx = blockIdx.x * blockDim.x + threadIdx.x;
    int e = idx >> 6;               // feature in low 6 bits
    if (e >= E) return;
    int f = idx & 63;
    int s = src[e], d = dst[e];
    float w = dinv[s] * dinv[d];
    atomicAdd(&acc[(size_t)d * D_H + f], w * h[(size_t)s * D_H + f]);
}

__global__ void k_bias_relu(float* __restrict__ acc, const float* __restrict__ b1, int N) {
    int idx = blockIdx.x * blockDim.x + threadIdx.x;
    if (idx < N * D_H) {
        int f = idx & 63;
        acc[idx] = fmaxf(acc[idx] + b1[f], 0.0f);
    }
}

// ---------------------------------------------------------------------------
// Self-loop init + edge scatter-add (layer 2, width 47, g padded to 48)
// ---------------------------------------------------------------------------
__global__ void k_self2(const float* __restrict__ dinv, const float* __restrict__ g,
                        float* __restrict__ xacc, int N) {
    int idx = blockIdx.x * blockDim.x + threadIdx.x;
    if (idx < N * D_O) {
        int i = idx / D_O;
        int c = idx - i * D_O;
        float w = dinv[i] * dinv[i];
        xacc[idx] = w * g[(size_t)i * D_OP + c];
    }
}

__global__ void k_scatter2(const int* __restrict__ src, const int* __restrict__ dst,
                           const float* __restrict__ dinv, const float* __restrict__ g,
                           float* __restrict__ xacc, int E) {
    long long idx = (long long)blockIdx.x * blockDim.x + threadIdx.x;
    long long tot = (long long)E * D_O;
    if (idx >= tot) return;
    int e = (int)(idx / D_O);
    int c = (int)(idx - (long long)e * D_O);
    int s = src[e], d = dst[e];
    float w = dinv[s] * dinv[d];
    atomicAdd(&xacc[(size_t)d * D_O + c], w * g[(size_t)s * D_OP + c]);
}

// ---------------------------------------------------------------------------
// Bias2 + softmax + argmax; xacc (in d_out) updated in place to xo = conv2 out
// ---------------------------------------------------------------------------
__global__ void k_finalize(const float* __restrict__ b2, float* __restrict__ xacc,
                           float* __restrict__ logits, float* __restrict__ preds, int N) {
    int i = blockIdx.x * blockDim.x + threadIdx.x;
    if (i >= N) return;
    float* xr = xacc + (size_t)i * D_O;
    float* lr = logits + (size_t)i * D_O;
    float mx = -INFINITY;
    int am = 0;
    for (int c = 0; c < D_O; ++c) {
        float v = xr[c] + b2[c];
        xr[c] = v;
        if (v > mx) { mx = v; am = c; }    // strict > : first-max like jnp.argmax
    }
    float sum = 0.0f;
    for (int c = 0; c < D_O; ++c) {
        float e = expf(xr[c] - mx);
        lr[c] = e;
        sum += e;
    }
    float inv = 1.0f / sum;
    for (int c = 0; c < D_O; ++c) lr[c] *= inv;
    preds[i] = (float)am;
}

// ---------------------------------------------------------------------------
extern "C" void kernel_launch(void* const* d_in, const int* in_sizes, int n_in,
                              void* d_out, int out_size, void* d_ws, size_t ws_size,
                              hipStream_t stream) {
    (void)n_in; (void)out_size; (void)ws_size;

    const float* x   = (const float*)d_in[0];   // [N, 256]
    const int*   ei  = (const int*)d_in[1];     // [2, E] row-major: src then dst
    const float* W1  = (const float*)d_in[2];   // [256, 64]
    const float* b1  = (const float*)d_in[3];   // [64]
    const float* W2  = (const float*)d_in[4];   // [64, 47]
    const float* b2  = (const float*)d_in[5];   // [47]

    const int N = in_sizes[0] / D_IN;           // 100000 (multiple of 16)
    const int E = in_sizes[1] / 2;              // 1600000
    const int* src = ei;
    const int* dst = ei + E;

    // workspace layout (floats)
    float* ws   = (float*)d_ws;
    float* dinv = ws;                           // N       (deg -> dinv in place)
    float* h1   = dinv + N;                     // N*64
    float* acc1 = h1 + (size_t)N * D_H;         // N*64    (scatter acc -> h2 in place)
    float* g    = acc1 + (size_t)N * D_H;       // N*48
    v2f*  Bp1   = (v2f*)(g + (size_t)N * D_OP); // (256/4)*4*32 v2f = 64 KB
    v2f*  Bp2   = Bp1 + (D_IN / 4) * 4 * 32;    // (64/4)*3*32  v2f = 12 KB

    // output layout (floats): logits [N*47] | preds [N] | xo [N*47]
    float* out_logits = (float*)d_out;
    float* out_preds  = out_logits + (size_t)N * D_O;
    float* out_xacc   = out_preds + N;

    const int B = 256;
    const int mtiles = N / 16;                      // 6250
    const int gemm_blocks = (mtiles + 7) / 8;       // 8 waves / block

    // 1. degree + rsqrt normalization
    k_deg_init<<<(N + B - 1) / B, B, 0, stream>>>(dinv, N);
    k_deg_acc<<<(E + B - 1) / B, B, 0, stream>>>(dst, dinv, E);
    k_deg_rsqrt<<<(N + B - 1) / B, B, 0, stream>>>(dinv, N);

    // 2. pack weights into WMMA lane order (zero-pads W2 col 47)
    {
        int t1 = (D_IN / 4) * 4 * 32;               // 8192 lanes
        int t2 = (D_H / 4) * 3 * 32;                // 1536 lanes
        k_packB<4><<<(t1 + B - 1) / B, B, 0, stream>>>(W1, Bp1, D_IN, D_H);
        k_packB<3><<<(t2 + B - 1) / B, B, 0, stream>>>(W2, Bp2, D_H, D_O);
    }

    // 3. layer 1: h1 = x @ W1   (WMMA fp32, 4 col-tiles, branch-free inner loop)
    k_gemm_wmma<4><<<gemm_blocks, B, 0, stream>>>(x, Bp1, h1, D_IN, D_H, mtiles);

    // 4. acc1 = dinv^2 * h1  (self loops), then scatter-add edges
    k_self1<<<((size_t)N * D_H + B - 1) / B, B, 0, stream>>>(dinv, h1, acc1, N);
    k_scatter1<<<(unsigned)(((size_t)E * D_H + B - 1) / B), B, 0, stream>>>(
        src, dst, dinv, h1, acc1, E);

    // 5. h2 = relu(acc1 + b1)  (in place)
    k_bias_relu<<<((size_t)N * D_H + B - 1) / B, B, 0, stream>>>(acc1, b1, N);

    // 6. layer 2: g = h2 @ W2  (WMMA fp32, 3 col-tiles, col 47 zero-padded)
    k_gemm_wmma<3><<<gemm_blocks, B, 0, stream>>>(acc1, Bp2, g, D_H, D_OP, mtiles);

    // 7. xo accumulation in d_out: self loops then edges
    k_self2<<<((size_t)N * D_O + B - 1) / B, B, 0, stream>>>(dinv, g, out_xacc, N);
    k_scatter2<<<(unsigned)(((size_t)E * D_O + B - 1) / B), B, 0, stream>>>(
        src, dst, dinv, g, out_xacc, E);

    // 8. bias2 + softmax + argmax
    k_finalize<<<(N + B - 1) / B, B, 0, stream>>>(b2, out_xacc, out_logits, out_preds, N);
}